// DBNBeatAndDownbeatDecoder_3032246911662
// MI455X (gfx1250) — compile-verified
//
#include <hip/hip_runtime.h>
#include <stdint.h>

// ---------------- problem constants (from reference) ----------------
#define B_        16
#define N_        3000
#define T_        42          // intervals 14..55
#define P_        220         // max bar length
#define TP_       (T_ * P_)   // 9240 states
#define MIN_INT   14
#define BLK       512         // 16 wave32 waves per WGP
#define NK        19          // ceil(9240 / 512)
#define NKP       (NK * BLK)  // 9728 padded states
#define CH        256         // frames staged per async chunk
#define NEGV      (-1.0e9f)
// sigmoid(x) >= 0.05  <=>  x >= log(0.05/0.95)
#define KEEP_LOGIT (-2.9444389791664403f)

// ---------------- gfx1250 async global->LDS path ----------------
#if defined(__has_builtin)
#if __has_builtin(__builtin_amdgcn_global_load_async_to_lds_b32)
#define HAVE_ASYNC_LDS 1
#endif
#endif

typedef __attribute__((address_space(1))) int gas_int;
typedef __attribute__((address_space(3))) int las_int;

__device__ __forceinline__ void async_cp_f32(const float* g, float* l) {
#ifdef HAVE_ASYNC_LDS
  __builtin_amdgcn_global_load_async_to_lds_b32((gas_int*)g, (las_int*)l, 0, 0);
#else
  *l = *g;  // fallback: plain LDS store
#endif
}

__device__ __forceinline__ void async_wait_all() {
#ifdef HAVE_ASYNC_LDS
  asm volatile("s_wait_asynccnt 0x0" ::: "memory");
#endif
}

__device__ __forceinline__ float lsig(float x) {
  // stable log_sigmoid with native exp/log
  return fminf(x, 0.0f) - __logf(1.0f + __expf(-fabsf(x)));
}

struct Res { float score; int t; int p; };

// =====================================================================
// Forward Viterbi: 1 workgroup per batch, double-buffered DP in LDS,
// one workgroup barrier per frame.
// =====================================================================
__global__ __launch_bounds__(BLK) void dbn_forward(
    const float* __restrict__ beat, const float* __restrict__ down,
    int* __restrict__ bps, Res* __restrict__ res)
{
  __shared__ float delta[2][NKP];     // 77824 B, double buffered
  __shared__ float trans[T_ * T_];    //  7056 B
  __shared__ float stage[2][2][CH];   //  4096 B double-buffered logits

  const int tid = threadIdx.x;
  const int b   = blockIdx.x;
  const float* gb = beat + (size_t)b * N_;
  const float* gd = down + (size_t)b * N_;

  // ---- per-thread state metadata, registers only ----
  // class: 1=beat, 2=none, 3=invalid/padding (write NEGV), 4=phase-0 (skip:
  //        written by lanes tid<T_ which own the tropical matvec)
  int cr[NK];
#pragma unroll
  for (int k = 0; k < NK; ++k) {
    int s = tid + k * BLK;
    int c = 3;
    if (s < TP_) {
      int t = s / P_;
      int p = s - t * P_;
      int iv = MIN_INT + t;
      if (p >= 4 * iv)        c = 3;
      else if (p == 0)        c = 4;
      else if ((p % iv) == 0) c = 1;
      else                    c = 2;
    }
    cr[k] = c;
  }

  // ---- transition matrix, row-normalized in double (startup only) ----
  if (tid < T_) {
    double lrt = log((double)(MIN_INT + tid));
    double sum = 0.0;
    for (int j = 0; j < T_; ++j)
      sum += exp(-100.0 * fabs(lrt - log((double)(MIN_INT + j))));
    double lz = log(sum);
    for (int j = 0; j < T_; ++j)
      trans[tid * T_ + j] =
          (float)(-100.0 * fabs(lrt - log((double)(MIN_INT + j))) - lz);
  }

  // ---- stage chunk 0 asynchronously: lanes 0..255 beat, 256..511 down ----
  {
    int i = tid & (CH - 1);
    int which = tid >> 8;
    if (i < N_) async_cp_f32((which ? gd : gb) + i, &stage[0][which][i]);
  }
  async_wait_all();
  __syncthreads();

  float* dcur = &delta[0][0];
  float* dnxt = &delta[1][0];

  // ---- frame 0: delta0 = emit(obs0) ----
  {
    float blv = stage[0][0][0];
    float dlv = stage[0][1][0];
    float o0 = lsig(dlv), o1 = lsig(blv), o2 = lsig(-fmaxf(blv, dlv));
    if (tid < T_) dcur[P_ * tid] = o0;          // phase-0 states
#pragma unroll
    for (int k = 0; k < NK; ++k) {
      int s = tid + k * BLK;
      if (cr[k] != 4)
        dcur[s] = (cr[k] == 3) ? NEGV : ((cr[k] == 1) ? o1 : o2);
    }
  }
  __syncthreads();

  int buf = 0;
  const int nch = (N_ + CH - 1) / CH;
  for (int ck = 0; ck < nch; ++ck) {
    // kick off async fill of the other staging buffer (overlaps the DP)
    if (ck + 1 < nch) {
      int i = tid & (CH - 1);
      int which = tid >> 8;
      int n = (ck + 1) * CH + i;
      if (n < N_) async_cp_f32((which ? gd : gb) + n, &stage[buf ^ 1][which][i]);
    }
    int n0 = ck * CH;
    int n1 = (n0 + CH < N_) ? (n0 + CH) : N_;
    for (int n = (ck == 0 ? 1 : n0); n < n1; ++n) {
      int i = n - n0;
      float blv = stage[buf][0][i];   // LDS broadcast reads
      float dlv = stage[buf][1][i];
      float o0 = lsig(dlv), o1 = lsig(blv), o2 = lsig(-fmaxf(blv, dlv));

      // ---- reads: phase shift sources (uniform, no divergence) ----
      float oldv[NK];
#pragma unroll
      for (int k = 0; k < NK; ++k) {
        int s = tid + k * BLK;
        int a = (s > 0) ? (s - 1) : 0;  // only tid0/k0 clamps (value unused)
        oldv[k] = dcur[a];
      }
      // ---- tropical matvec + backpointer on lanes 0..41 ----
      float m = NEGV;
      int bpv = 0;
      if (tid < T_) {
        for (int j = 0; j < T_; ++j) {
          float v = dcur[224 * j + 55] + trans[j * T_ + tid];
          if (v > m) { m = v; bpv = j; }   // first-max semantics
        }
      }

      // ---- writes into the other buffer ----
      if (tid < T_) {
        bps[((size_t)b * (N_ - 1) + (n - 1)) * T_ + tid] = bpv;
        dnxt[P_ * tid] = m + o0;           // phase-0 states (class 4)
      }
#pragma unroll
      for (int k = 0; k < NK; ++k) {
        int s = tid + k * BLK;
        if (cr[k] != 4) {
          float e = (cr[k] == 1) ? o1 : o2;
          dnxt[s] = (cr[k] == 3) ? NEGV : (oldv[k] + e);
        }
      }
      __syncthreads();                      // single barrier per frame
      float* tmp = dcur; dcur = dnxt; dnxt = tmp;
    }
    if (ck + 1 < nch) {
      async_wait_all();
      __syncthreads();
    }
    buf ^= 1;
  }

  // ---- final argmax over flat states (tie -> smaller flat index) ----
  float bestv = NEGV - 1.0f;
  int besti = 0;
#pragma unroll
  for (int k = 0; k < NK; ++k) {
    int s = tid + k * BLK;
    if (s < TP_) {
      float v = dcur[s];
      if (v > bestv) { bestv = v; besti = s; }
    }
  }
  float* red_v = &stage[0][0][0];        // reuse staging LDS as scratch
  int*   red_i = (int*)&stage[1][0][0];
  red_v[tid] = bestv;
  red_i[tid] = besti;
  __syncthreads();
  for (int off = BLK / 2; off > 0; off >>= 1) {
    if (tid < off) {
      float v2 = red_v[tid + off];
      int   i2 = red_i[tid + off];
      if (v2 > red_v[tid] || (v2 == red_v[tid] && i2 < red_i[tid])) {
        red_v[tid] = v2; red_i[tid] = i2;
      }
    }
    __syncthreads();
  }
  if (tid == 0) {
    int s = red_i[0];
    int t = s / P_;
    res[b].score = red_v[0];
    res[b].t = t;
    res[b].p = s - t * P_;
  }
}

// =====================================================================
// Backtrace + mask emission: one lane per batch.
// =====================================================================
__global__ void dbn_backtrack(const float* __restrict__ beat,
                              const float* __restrict__ down,
                              const int* __restrict__ bps,
                              const Res* __restrict__ res,
                              float* __restrict__ out)
{
  int b = threadIdx.x;
  if (b >= B_) return;
  int t = res[b].t;
  int p = res[b].p;
  out[2 * B_ * N_ + b] = res[b].score;
  for (int n = N_ - 1; n >= 0; --n) {
    if (n >= 8) __builtin_prefetch(&beat[(size_t)b * N_ + n - 8], 0, 0);
    float blv = beat[(size_t)b * N_ + n];
    float dlv = down[(size_t)b * N_ + n];
    bool keep = fmaxf(blv, dlv) >= KEEP_LOGIT;   // sigmoid monotonic
    int iv = MIN_INT + t;
    out[(size_t)b * N_ + n]            = (((p % iv) == 0) && keep) ? 1.0f : 0.0f;
    out[(size_t)B_ * N_ + b * N_ + n]  = ((p == 0) && keep)        ? 1.0f : 0.0f;
    if (n > 0) {
      if (p == 0) {
        int tp = bps[((size_t)b * (N_ - 1) + (n - 1)) * T_ + t];
        t = tp;
        p = 4 * (MIN_INT + tp) - 1;
      } else {
        --p;
      }
    }
  }
}

// =====================================================================
extern "C" void kernel_launch(void* const* d_in, const int* in_sizes, int n_in,
                              void* d_out, int out_size, void* d_ws, size_t ws_size,
                              hipStream_t stream) {
  const float* beat = (const float*)d_in[0];
  const float* down = (const float*)d_in[1];
  float* out = (float*)d_out;

  // workspace: backpointers [B][N-1][T] int32 (~8.06 MB), then per-batch Res
  int* bps = (int*)d_ws;
  size_t bps_bytes = (size_t)B_ * (N_ - 1) * T_ * sizeof(int);
  Res* res = (Res*)((char*)d_ws + ((bps_bytes + 255) & ~(size_t)255));

  dbn_forward<<<B_, BLK, 0, stream>>>(beat, down, bps, res);
  dbn_backtrack<<<1, 32, 0, stream>>>(beat, down, bps, res, out);
}